// HardBatchedTripletLoss_89103391523252
// MI455X (gfx1250) — compile-verified
//
#include <hip/hip_runtime.h>
#include <hip/hip_bf16.h>

typedef __attribute__((ext_vector_type(2))) float v2f;
typedef __attribute__((ext_vector_type(4))) float v4f;
typedef __attribute__((ext_vector_type(8))) float v8f;

#define MARGIN 0.25f

// ---------------------------------------------------------------------------
// Row squared-norm: one wave per row of 128 floats.
// ---------------------------------------------------------------------------
__global__ __launch_bounds__(256) void rownorm_kernel(
    const float* __restrict__ src, float* __restrict__ dst, int nrows) {
  const int lane = threadIdx.x & 31;
  const int wave = threadIdx.x >> 5;
  const int row  = blockIdx.x * 8 + wave;
  if (row >= nrows) return;
  v4f v = *(const v4f*)(src + (size_t)row * 128 + lane * 4);
  float s = v.x * v.x + v.y * v.y + v.z * v.z + v.w * v.w;
  s += __shfl_xor(s, 1);
  s += __shfl_xor(s, 2);
  s += __shfl_xor(s, 4);
  s += __shfl_xor(s, 8);
  s += __shfl_xor(s, 16);
  if (lane == 0) dst[row] = s;
}

// ---------------------------------------------------------------------------
// Reset reduction accumulators (workspace is not re-initialized by harness).
// ---------------------------------------------------------------------------
__global__ void init_kernel(unsigned int* __restrict__ dpos_bits,
                            unsigned int* __restrict__ dneg_bits, int M) {
  const int i = blockIdx.x * blockDim.x + threadIdx.x;
  if (i < M) dpos_bits[i] = 0u;                 // +0.0f, <= any distance
  if (i == 0) dneg_bits[0] = 0x7F800000u;       // +inf
}

// ---------------------------------------------------------------------------
// WMMA distance kernel.
//   dot = A[16,128] x B_cols[16,128]^T  via 32x V_WMMA_F32_16X16X4_F32
//   dist = sqrt(max(|a|^2 + |b|^2 - 2*dot, 0))
// Fragment layout (ISA 7.12.2): lane l holds (row = l%16, K = 2*(l>=16)+{0,1})
// for both A and B -> one aligned b64 load per fragment.
// MODE 0: per-anchor-row max -> atomicMax on uint bits.
// MODE 1: global min        -> atomicMin on uint bits.
// Each wave: one 16-row anchor block, 8 column tiles (128 cols).
// ---------------------------------------------------------------------------
template <int MODE>
__global__ __launch_bounds__(256) void dist_kernel(
    const float* __restrict__ Arows,    // [M,128] anchors
    const float* __restrict__ Brows,    // [ncols,128] positives or negatives
    const float* __restrict__ anorm,    // [M]
    const float* __restrict__ bnorm,    // [ncols]
    unsigned int* __restrict__ row_out, // [M] bits (MODE 0)
    unsigned int* __restrict__ g_out)   // [1] bits (MODE 1)
{
  const int lane = threadIdx.x & 31;
  const int wave = threadIdx.x >> 5;
  const int half = lane >> 4;   // 0: K 0..1 / rows 0..7, 1: K 2..3 / rows 8..15
  const int r    = lane & 15;

  const int m0     = (blockIdx.x * 8 + wave) * 16;
  const int n0base = blockIdx.y * 128;

  // Preload the 32 A fragments covering K = 0..127 for this 16-row block.
  v2f afrag[32];
  const float* aptr = Arows + (size_t)(m0 + r) * 128 + 2 * half;
#pragma unroll
  for (int kk = 0; kk < 32; ++kk)
    afrag[kk] = *(const v2f*)(aptr + kk * 4);

  // Anchor norms for this lane's 8 C-row slots (C VGPR i -> row i + 8*half).
  float am[8];
#pragma unroll
  for (int i = 0; i < 8; ++i)
    am[i] = anorm[m0 + i + 8 * half];

  float rowmax[8];
#pragma unroll
  for (int i = 0; i < 8; ++i) rowmax[i] = -1.0f;
  float gmin = 3.0e38f;

  for (int t = 0; t < 8; ++t) {
    const int n0 = n0base + t * 16;
    const float* bptr = Brows + (size_t)(n0 + r) * 128 + 2 * half;

    v2f bfrag[32];
#pragma unroll
    for (int kk = 0; kk < 32; ++kk)
      bfrag[kk] = *(const v2f*)(bptr + kk * 4);

    v8f c = {0.f, 0.f, 0.f, 0.f, 0.f, 0.f, 0.f, 0.f};
#pragma unroll
    for (int kk = 0; kk < 32; ++kk)
      c = __builtin_amdgcn_wmma_f32_16x16x4_f32(
          /*neg_a=*/false, afrag[kk], /*neg_b=*/false, bfrag[kk],
          /*c_mod=*/(short)0, c, /*reuse_a=*/false, /*reuse_b=*/false);

    const float bn = bnorm[n0 + r];  // C element N index == r for all 8 VGPRs
#pragma unroll
    for (int i = 0; i < 8; ++i) {
      float sq = am[i] + bn - 2.0f * c[i];
      float d  = __builtin_sqrtf(fmaxf(sq, 0.0f));
      if (MODE == 0) rowmax[i] = fmaxf(rowmax[i], d);
      else           gmin      = fminf(gmin, d);
    }
  }

  if (MODE == 0) {
    // Row-wise max: reduce over the 16 lanes of each half (masks < 16 stay
    // inside a half), then one atomic per row.
#pragma unroll
    for (int i = 0; i < 8; ++i) {
      float v = rowmax[i];
      v = fmaxf(v, __shfl_xor(v, 1));
      v = fmaxf(v, __shfl_xor(v, 2));
      v = fmaxf(v, __shfl_xor(v, 4));
      v = fmaxf(v, __shfl_xor(v, 8));
      if (r == 0)
        atomicMax(&row_out[m0 + i + 8 * half], __float_as_uint(v));
    }
  } else {
    gmin = fminf(gmin, __shfl_xor(gmin, 1));
    gmin = fminf(gmin, __shfl_xor(gmin, 2));
    gmin = fminf(gmin, __shfl_xor(gmin, 4));
    gmin = fminf(gmin, __shfl_xor(gmin, 8));
    gmin = fminf(gmin, __shfl_xor(gmin, 16));
    if (lane == 0)
      atomicMin(g_out, __float_as_uint(gmin));
  }
}

// ---------------------------------------------------------------------------
// losses.mean(): relu(d_pos[b] - d_neg + margin), mean over B.
// ---------------------------------------------------------------------------
__global__ __launch_bounds__(256) void finalize_kernel(
    const unsigned int* __restrict__ dpos_bits,
    const unsigned int* __restrict__ dneg_bits,
    float* __restrict__ out, int M) {
  __shared__ float sm[256];
  const float dneg = __uint_as_float(dneg_bits[0]);
  float s = 0.0f;
  for (int b = threadIdx.x; b < M; b += 256) {
    float l = __uint_as_float(dpos_bits[b]) - dneg + MARGIN;
    s += (l > 0.0f) ? l : 0.0f;
  }
  sm[threadIdx.x] = s;
  __syncthreads();
  for (int off = 128; off > 0; off >>= 1) {
    if ((int)threadIdx.x < off) sm[threadIdx.x] += sm[threadIdx.x + off];
    __syncthreads();
  }
  if (threadIdx.x == 0) out[0] = sm[0] / (float)M;
}

// ---------------------------------------------------------------------------
extern "C" void kernel_launch(void* const* d_in, const int* in_sizes, int n_in,
                              void* d_out, int out_size, void* d_ws, size_t ws_size,
                              hipStream_t stream) {
  const float* anchor   = (const float*)d_in[0];
  const float* positive = (const float*)d_in[1];
  const float* negative = (const float*)d_in[2];

  const int M  = in_sizes[0] / 128;  // 2048 anchors
  const int P  = in_sizes[1] / 128;  // 2048 positives
  const int NN = in_sizes[2] / 128;  // 32768 flattened negatives

  char* ws = (char*)d_ws;
  float* anorm = (float*)ws;            ws += (size_t)M  * 4;
  float* pnorm = (float*)ws;            ws += (size_t)P  * 4;
  float* nnorm = (float*)ws;            ws += (size_t)NN * 4;
  unsigned int* dpos = (unsigned int*)ws; ws += (size_t)M * 4;
  unsigned int* dneg = (unsigned int*)ws;

  rownorm_kernel<<<(M + 7) / 8, 256, 0, stream>>>(anchor, anorm, M);
  rownorm_kernel<<<(P + 7) / 8, 256, 0, stream>>>(positive, pnorm, P);
  rownorm_kernel<<<(NN + 7) / 8, 256, 0, stream>>>(negative, nnorm, NN);
  init_kernel<<<(M + 255) / 256, 256, 0, stream>>>(dpos, dneg, M);

  // d_pos: anchors x positives, row-max.  grid = (M/128 wave-groups, P/128 col chunks)
  dim3 gpos(M / 128, P / 128);
  dist_kernel<0><<<gpos, 256, 0, stream>>>(anchor, positive, anorm, pnorm, dpos, nullptr);

  // d_neg: anchors x negatives, global min.
  dim3 gneg(M / 128, NN / 128);
  dist_kernel<1><<<gneg, 256, 0, stream>>>(anchor, negative, anorm, nnorm, nullptr, dneg);

  finalize_kernel<<<1, 256, 0, stream>>>(dpos, dneg, (float*)d_out, M);
}